// PoolingLayer_12008728559998
// MI455X (gfx1250) — compile-verified
//
#include <hip/hip_runtime.h>
#include <stdint.h>

#define BATCH 4
#define NROWS 65536
#define DIM   256
#define KPOOL 16384

typedef float v2f __attribute__((ext_vector_type(2)));
typedef float v8f __attribute__((ext_vector_type(8)));

// ---------------------------------------------------------------------------
// Pass 1: per-row sum of squares  +  per-block partial column sums.
// One block = 256 rows, 8 waves x 32 rows; each wave streams a full 1KB row
// (lane reads 8 contiguous floats). Deterministic fixed-order reductions.
// ---------------------------------------------------------------------------
__global__ __launch_bounds__(256) void k1_sums(const float* __restrict__ x,
                                               float* __restrict__ rowsq,
                                               float* __restrict__ partial) {
    __shared__ float scol[8 * 256];
    const int b    = blockIdx.x >> 8;     // 256 blocks per batch
    const int blk  = blockIdx.x & 255;
    const int t    = threadIdx.x;
    const int w    = t >> 5;
    const int lane = t & 31;

    const float* xb = x + (size_t)b * NROWS * DIM;
    float ca[8];
#pragma unroll
    for (int j = 0; j < 8; ++j) ca[j] = 0.0f;

    const int rowBase = blk * 256 + w * 32;
    for (int i = 0; i < 32; ++i) {
        const int row = rowBase + i;
        const float4* p = (const float4*)(xb + (size_t)row * DIM + lane * 8);
        float4 v0 = p[0];
        float4 v1 = p[1];
        float sq = v0.x * v0.x + v0.y * v0.y + v0.z * v0.z + v0.w * v0.w +
                   v1.x * v1.x + v1.y * v1.y + v1.z * v1.z + v1.w * v1.w;
        ca[0] += v0.x; ca[1] += v0.y; ca[2] += v0.z; ca[3] += v0.w;
        ca[4] += v1.x; ca[5] += v1.y; ca[6] += v1.z; ca[7] += v1.w;
#pragma unroll
        for (int m = 16; m >= 1; m >>= 1) sq += __shfl_xor(sq, m, 32);
        if (lane == 0) rowsq[(size_t)b * NROWS + row] = sq;
    }
#pragma unroll
    for (int j = 0; j < 8; ++j) scol[w * 256 + lane * 8 + j] = ca[j];
    __syncthreads();
    float ssum = 0.0f;            // fixed-order (deterministic) wave reduction
#pragma unroll
    for (int w2 = 0; w2 < 8; ++w2) ssum += scol[w2 * 256 + t];
    partial[((size_t)b * 256 + blk) * 256 + t] = ssum;
}

// ---------------------------------------------------------------------------
// Pass 2: reduce 256 per-block partials -> scaled column sums (2/N)*S[b][d]
// ---------------------------------------------------------------------------
__global__ __launch_bounds__(256) void k2_reduce(const float* __restrict__ partial,
                                                 float* __restrict__ Scol) {
    const int b = blockIdx.x;
    const int t = threadIdx.x;
    float s = 0.0f;
    for (int k = 0; k < 256; ++k) s += partial[((size_t)b * 256 + k) * 256 + t];
    Scol[b * 256 + t] = s * (2.0f / (float)NROWS);
}

// ---------------------------------------------------------------------------
// Pass 3: key[b][n] = ||x_n||^2 - (2/N) * dot(x_n, S_b)   via WMMA f32 16x16x4.
// Block = 128 rows (8 waves x 16-row tiles). x tile staged through padded LDS
// (stride 65 -> conflict-free column reads). B matrix = scaled S slice
// replicated across N columns, so D[:,0] holds the 16 dots.
// ---------------------------------------------------------------------------
__global__ __launch_bounds__(256) void k3_keys(const float* __restrict__ x,
                                               const float* __restrict__ Scol,
                                               const float* __restrict__ rowsq,
                                               float* __restrict__ keys) {
    __shared__ float Sv[256];
    __shared__ float tile[128 * 65];
    const int b  = blockIdx.x >> 9;       // 512 blocks per batch
    const int tb = blockIdx.x & 511;
    const int t    = threadIdx.x;
    const int w    = t >> 5;
    const int lane = t & 31;
    const int rowBase = tb * 128;

    Sv[t] = Scol[b * 256 + t];            // already scaled by 2/N

    const float* xb = x + ((size_t)b * NROWS + rowBase) * DIM;
    const int m  = lane & 15;             // A-matrix row within the tile
    const int ko = (lane >> 4) << 1;      // K sub-offset per ISA A layout

    v8f acc = {0.f, 0.f, 0.f, 0.f, 0.f, 0.f, 0.f, 0.f};

    for (int cc = 0; cc < 4; ++cc) {      // 4 column chunks of 64
        __syncthreads();
        const int rsub = t >> 4;
        const int c4   = t & 15;
#pragma unroll
        for (int it = 0; it < 8; ++it) {  // cooperative coalesced tile load
            const int row = it * 16 + rsub;
            const float4* p = (const float4*)(xb + (size_t)row * DIM + cc * 64);
            float4 v = p[c4];
            float* dst = &tile[row * 65 + c4 * 4];
            dst[0] = v.x; dst[1] = v.y; dst[2] = v.z; dst[3] = v.w;
        }
        __syncthreads();
        const float* ta = &tile[(w * 16 + m) * 65];
        const float* sb = &Sv[cc * 64];
#pragma unroll
        for (int k0 = 0; k0 < 64; k0 += 4) {
            v2f a, bb;
            a.x  = ta[k0 + ko];
            a.y  = ta[k0 + ko + 1];
            bb.x = sb[k0 + ko];
            bb.y = sb[k0 + ko + 1];
            acc = __builtin_amdgcn_wmma_f32_16x16x4_f32(
                false, a, false, bb, (short)0, acc, false, false);
        }
    }

    // D column 0: lane 0 holds rows 0..7 of the wave tile, lane 16 rows 8..15
    if ((lane & 15) == 0) {
        const int half = lane >> 4;
        const size_t g0 = (size_t)b * NROWS + rowBase + w * 16 + half * 8;
#pragma unroll
        for (int i = 0; i < 8; ++i)
            keys[g0 + i] = rowsq[g0 + i] - acc[i];
    }
}

// ---------------------------------------------------------------------------
// Pass 4: per-batch stable LSD radix sort (8 x 4-bit) of inverted order-keys
// with index payload -> descending by r, ties ascending index (jax top_k).
// One 512-thread block per batch, 128 items/thread, 32KB LDS histogram.
// ---------------------------------------------------------------------------
#define SORT_T 512
#define CHUNK  128

__global__ __launch_bounds__(512) void k4_sort(const float* __restrict__ keys,
                                               uint64_t* __restrict__ buf0,
                                               uint64_t* __restrict__ buf1) {
    __shared__ uint32_t hist[16 * SORT_T];
    __shared__ uint32_t totals[16];
    __shared__ uint32_t basev[16];
    const int b = blockIdx.x;
    const int t = threadIdx.x;
    const size_t base = (size_t)b * NROWS;

    for (int n = t; n < NROWS; n += SORT_T) {
        const uint32_t u   = __float_as_uint(keys[base + n]);
        const uint32_t asc = (u & 0x80000000u) ? ~u : (u | 0x80000000u);
        buf0[base + n] = ((uint64_t)(~asc) << 32) | (uint32_t)n;  // sort ascending
    }
    __threadfence();
    __syncthreads();

    uint64_t* src = buf0;
    uint64_t* dst = buf1;
    for (int p = 0; p < 8; ++p) {
        const int shift = p * 4;
        uint32_t cnt[16];
#pragma unroll
        for (int d = 0; d < 16; ++d) cnt[d] = 0;
        const size_t cbase = base + (size_t)t * CHUNK;
        for (int i = 0; i < CHUNK; ++i) {
            const uint32_t hi = (uint32_t)(src[cbase + i] >> 32);
            cnt[(hi >> shift) & 15]++;
        }
#pragma unroll
        for (int d = 0; d < 16; ++d) hist[d * SORT_T + t] = cnt[d];
        __syncthreads();

        const int w = t >> 5, lane = t & 31;
        if (lane == 0) {                  // 16 waves: one digit column each
            uint32_t run = 0;
            for (int i = 0; i < SORT_T; ++i) {
                const uint32_t c = hist[w * SORT_T + i];
                hist[w * SORT_T + i] = run;
                run += c;
            }
            totals[w] = run;
        }
        __syncthreads();
        if (t == 0) {
            uint32_t run = 0;
            for (int d = 0; d < 16; ++d) { basev[d] = run; run += totals[d]; }
        }
        __syncthreads();

        uint32_t off[16];
#pragma unroll
        for (int d = 0; d < 16; ++d) off[d] = hist[d * SORT_T + t] + basev[d];
        for (int i = 0; i < CHUNK; ++i) {
            const uint64_t v  = src[cbase + i];
            const uint32_t dd = ((uint32_t)(v >> 32) >> shift) & 15;
            dst[base + (off[dd]++)] = v;
        }
        __threadfence();
        __syncthreads();
        uint64_t* tmp = src; src = dst; dst = tmp;
    }
    // 8 passes: result lands back in buf0
}

// ---------------------------------------------------------------------------
// Pass 5: gather kept rows (wave per 1KB row) + mask as float after x block.
// ---------------------------------------------------------------------------
__global__ __launch_bounds__(256) void k5_gather(const float* __restrict__ x,
                                                 const unsigned char* __restrict__ mask,
                                                 const uint64_t* __restrict__ sorted,
                                                 float* __restrict__ out) {
    const int t    = threadIdx.x;
    const int w    = t >> 5;
    const int lane = t & 31;
    const size_t g = (size_t)blockIdx.x * 8 + w;      // kept-row id in [0, B*K)
    const int b = (int)(g >> 14);                     // / 16384
    const int k = (int)(g & (KPOOL - 1));
    const uint32_t idx = (uint32_t)sorted[(size_t)b * NROWS + k];

    const float4* p = (const float4*)(x + ((size_t)b * NROWS + idx) * DIM + lane * 8);
    float4* q = (float4*)(out + g * DIM + lane * 8);
    q[0] = p[0];
    q[1] = p[1];
    if (lane == 0)
        out[(size_t)BATCH * KPOOL * DIM + g] = (float)mask[(size_t)b * NROWS + idx];
}

// ---------------------------------------------------------------------------
extern "C" void kernel_launch(void* const* d_in, const int* in_sizes, int n_in,
                              void* d_out, int out_size, void* d_ws, size_t ws_size,
                              hipStream_t stream) {
    (void)in_sizes; (void)n_in; (void)out_size; (void)ws_size;
    const float*         x    = (const float*)d_in[0];
    const unsigned char* mask = (const unsigned char*)d_in[1];
    float*               out  = (float*)d_out;
    char*                ws   = (char*)d_ws;

    float*    rowsq   = (float*)(ws + 0);                         // 1 MB
    float*    partial = (float*)(ws + (1u << 20));                // 1 MB
    float*    Scol    = (float*)(ws + (2u << 20));                // 4 KB
    float*    keys    = (float*)(ws + (1u << 20));                // reuses partial
    uint64_t* buf0    = (uint64_t*)(ws + (2u << 20) + 65536);     // 2 MB
    uint64_t* buf1    = (uint64_t*)(ws + (4u << 20) + 65536);     // 2 MB

    k1_sums  <<<dim3(BATCH * 256),        dim3(256), 0, stream>>>(x, rowsq, partial);
    k2_reduce<<<dim3(BATCH),              dim3(256), 0, stream>>>(partial, Scol);
    k3_keys  <<<dim3(BATCH * 512),        dim3(256), 0, stream>>>(x, Scol, rowsq, keys);
    k4_sort  <<<dim3(BATCH),              dim3(512), 0, stream>>>(keys, buf0, buf1);
    k5_gather<<<dim3(BATCH * KPOOL / 8),  dim3(256), 0, stream>>>(x, mask, buf0, out);
}